// VectorNormSelection_48137993454061
// MI455X (gfx1250) — compile-verified
//
#include <hip/hip_runtime.h>
#include <stdint.h>

// Top-16-of-64 squared-norm selection, bit-exact vs jax.lax.top_k ordering.
//
// Layout: x[N][64][3] f32 -> out[N][16][3] f32.
// One wave processes 2 rows per stage; async global->LDS double buffering.
// Streaming data is marked non-temporal: x (384MB) and out (96MB) are each
// touched exactly once, so letting them occupy the 192MB L2 is pure pollution.

#define BLOCK            256
#define WAVES_PER_BLOCK  8        // wave32
#define ROW_FLOATS       192
#define PAIR_BYTES       1536     // 2 rows * 192 f32
#define KEY_BYTES        512      // 64 keys * 8B
#define WAVE_LDS         (2 * PAIR_BYTES + KEY_BYTES)   // 3584 B / wave

// ---- CDNA5 async global->LDS (ASYNCcnt), non-temporal ------------------
__device__ __forceinline__ void async_copy16(uint32_t lds_addr, uint64_t gaddr) {
  // LDS address comes from the vdst VGPR; per-lane 16B chunk. NT: streamed once.
  asm volatile("global_load_async_to_lds_b128 %0, %1, off th:TH_LOAD_NT"
               :: "v"(lds_addr), "v"(gaddr)
               : "memory");
}
__device__ __forceinline__ void wait_async_le3() {
  asm volatile("s_wait_asynccnt 0x3" ::: "memory");
}
__device__ __forceinline__ void wait_ds0() {
  asm volatile("s_wait_dscnt 0x0" ::: "memory");
}

// Exact (non-FMA) squared norm so ordering matches jnp.sum(x*x) bit-for-bit.
__device__ __forceinline__ float sqnorm(float x, float y, float z) {
  return __fadd_rn(__fadd_rn(__fmul_rn(x, x), __fmul_rn(y, y)), __fmul_rn(z, z));
}

__device__ __forceinline__ void store3_nt(float* o, float a, float b, float c) {
  __builtin_nontemporal_store(a, o + 0);
  __builtin_nontemporal_store(b, o + 1);
  __builtin_nontemporal_store(c, o + 2);
}

__device__ __forceinline__ void rank_and_emit(
    const unsigned long long* keys, unsigned long long kA, unsigned long long kB,
    float ax, float ay, float az, float bx, float by, float bz,
    float* __restrict__ orow) {
  unsigned ca = 0, cb = 0;
  #pragma unroll 8
  for (int j = 0; j < 64; ++j) {
    unsigned long long kj = keys[j];     // ds broadcast load, co-issues w/ VALU
    ca += (kj > kA);                     // single v_cmp_gt_u64 + carry add
    cb += (kj > kB);
  }
  if (ca < 16u) store3_nt(orow + ca * 3, ax, ay, az);
  if (cb < 16u) store3_nt(orow + cb * 3, bx, by, bz);
}

__global__ __launch_bounds__(BLOCK) void
VectorNormSelection_48137993454061_kernel(const float* __restrict__ x,
                                          float* __restrict__ out,
                                          int nPairs, int oddRow) {
  __shared__ __align__(16) unsigned char smem[WAVES_PER_BLOCK * WAVE_LDS];
  const int lane  = threadIdx.x & 31;
  const int wslot = threadIdx.x >> 5;
  unsigned char* wbase = smem + wslot * WAVE_LDS;
  unsigned long long* keys = (unsigned long long*)(wbase + 2 * PAIR_BYTES);

  const int wid = blockIdx.x * WAVES_PER_BLOCK + wslot;
  const int nW  = gridDim.x * WAVES_PER_BLOCK;

  // Robustness: odd trailing row handled directly from global by wave 0.
  if (oddRow >= 0 && wid == 0) {
    const float* rowp = x + (size_t)oddRow * ROW_FLOATS;
    float ax = rowp[lane * 3 + 0],  ay = rowp[lane * 3 + 1],  az = rowp[lane * 3 + 2];
    float bx = rowp[96 + lane * 3], by = rowp[97 + lane * 3], bz = rowp[98 + lane * 3];
    float na = sqnorm(ax, ay, az), nb = sqnorm(bx, by, bz);
    unsigned long long kA =
        ((unsigned long long)__float_as_uint(na) << 32) | (unsigned)(63 - lane);
    unsigned long long kB =
        ((unsigned long long)__float_as_uint(nb) << 32) | (unsigned)(31 - lane);
    keys[lane] = kA; keys[lane + 32] = kB;
    rank_and_emit(keys, kA, kB, ax, ay, az, bx, by, bz,
                  out + (size_t)oddRow * 48);
  }
  if (wid >= nPairs) return;

  // Prefetch first row pair into buffer 0 (3 full-wave b128 async loads).
  {
    uint64_t g  = (uint64_t)(uintptr_t)x + (uint64_t)wid * PAIR_BYTES;
    uint32_t l0 = (uint32_t)(uintptr_t)wbase;
    #pragma unroll
    for (int q = 0; q < 3; ++q)
      async_copy16(l0 + (uint32_t)(q * 512 + lane * 16),
                   g + (uint64_t)(q * 512 + lane * 16));
  }

  int cur = 0;
  for (int p = wid; p < nPairs; p += nW) {
    int pn = p + nW;
    if (pn >= nPairs) pn = p;   // harmless re-load; data unused

    // All our ds reads of the other buffer must drain before async rewrites it.
    wait_ds0();
    {
      uint64_t g  = (uint64_t)(uintptr_t)x + (uint64_t)pn * PAIR_BYTES;
      uint32_t l0 = (uint32_t)(uintptr_t)(wbase + (cur ^ 1) * PAIR_BYTES);
      #pragma unroll
      for (int q = 0; q < 3; ++q)
        async_copy16(l0 + (uint32_t)(q * 512 + lane * 16),
                     g + (uint64_t)(q * 512 + lane * 16));
    }
    // <=3 outstanding: the 3 loads of the *current* buffer have completed
    // (async loads complete in order).
    wait_async_le3();

    const float* buf = (const float*)(wbase + cur * PAIR_BYTES);
    #pragma unroll
    for (int r = 0; r < 2; ++r) {
      const float* rowp = buf + r * ROW_FLOATS;
      // Lane owns vectors (lane) and (lane+32) of this row.
      float ax = rowp[lane * 3 + 0],  ay = rowp[lane * 3 + 1],  az = rowp[lane * 3 + 2];
      float bx = rowp[96 + lane * 3], by = rowp[97 + lane * 3], bz = rowp[98 + lane * 3];
      float na = sqnorm(ax, ay, az), nb = sqnorm(bx, by, bz);
      // Unique sortable key: norm bits (f32>=0 is order-isomorphic as u32),
      // tie-broken by smaller original index first (top_k stability).
      unsigned long long kA =
          ((unsigned long long)__float_as_uint(na) << 32) | (unsigned)(63 - lane);
      unsigned long long kB =
          ((unsigned long long)__float_as_uint(nb) << 32) | (unsigned)(31 - lane);
      keys[lane] = kA;            // same-wave LDS pipe is in-order: no barrier
      keys[lane + 32] = kB;
      rank_and_emit(keys, kA, kB, ax, ay, az, bx, by, bz,
                    out + (size_t)(2 * p + r) * 48);
    }
    cur ^= 1;
  }
}

extern "C" void kernel_launch(void* const* d_in, const int* in_sizes, int n_in,
                              void* d_out, int out_size, void* d_ws, size_t ws_size,
                              hipStream_t stream) {
  (void)n_in; (void)out_size; (void)d_ws; (void)ws_size;
  const float* x = (const float*)d_in[0];
  float* out = (float*)d_out;

  long long nRows = (long long)in_sizes[0] / ROW_FLOATS;   // 500000
  long long pairs = nRows >> 1;
  int oddRow = (nRows & 1) ? (int)(nRows - 1) : -1;

  long long blocksNeeded = (pairs + WAVES_PER_BLOCK - 1) / WAVES_PER_BLOCK;
  int grid = (int)(blocksNeeded < 2048 ? (blocksNeeded > 0 ? blocksNeeded : 1)
                                       : 2048);
  VectorNormSelection_48137993454061_kernel<<<grid, BLOCK, 0, stream>>>(
      x, out, (int)pairs, oddRow);
}